// MultiHeadAttention_31172872634849
// MI455X (gfx1250) — compile-verified
//
#include <hip/hip_runtime.h>
#include <math.h>

// Problem constants (from reference)
#define BS   2
#define N    2048
#define DIN  128
#define DOUT 128
#define H    8
#define DK   16
#define SROW 2050   // padded LDS row stride (dwords): rows land on distinct banks, 8B-aligned

typedef __attribute__((ext_vector_type(2))) float v2f;
typedef __attribute__((ext_vector_type(8))) float v8f;

// Full-precision fp32 WMMA: D(16x16) = A(16x4) * B(4x16) + C
static __device__ __forceinline__ v8f wmma4(v2f a, v2f b, v8f c) {
  return __builtin_amdgcn_wmma_f32_16x16x4_f32(
      /*neg_a=*/false, a, /*neg_b=*/false, b,
      /*c_mod=*/(short)0, c, /*reuse_a=*/false, /*reuse_b=*/false);
}

// ---------------------------------------------------------------------------
// Kernel 1: QKV projection.  y[g][o] = sum_i x[g][i]*W[o][i] + b[o]
// stored per-head: P[((b*H + o/16)*N + n)*16 + o%16]
// grid = BS*N/16 blocks (16 rows each), 256 threads (8 waves, one 16-col tile each)
// ---------------------------------------------------------------------------
__global__ void qkv_proj_kernel(const float* __restrict__ x,
                                const float* __restrict__ Wq, const float* __restrict__ bq,
                                const float* __restrict__ Wk, const float* __restrict__ bk,
                                const float* __restrict__ Wv, const float* __restrict__ bv,
                                float* __restrict__ Q, float* __restrict__ Kd,
                                float* __restrict__ V) {
  const int lane = threadIdx.x & 31;
  const int wave = threadIdx.x >> 5;
  const int l16  = lane & 15;
  const int hl   = lane >> 4;            // half-wave select (K-pair 0/1 within frag)
  const int rowBase = blockIdx.x * 16;   // global row over BS*N
  const int b  = rowBase / N;
  const int n0 = rowBase % N;
  const int oBase = wave * 16;

  // Preload all 32 A-fragments of x (reused for Wq, Wk, Wv)
  v2f a[32];
#pragma unroll
  for (int kk = 0; kk < 32; ++kk)
    a[kk] = *(const v2f*)(x + (size_t)(rowBase + l16) * DIN + kk * 4 + hl * 2);

  const float* Ws[3]  = {Wq, Wk, Wv};
  const float* Bs[3]  = {bq, bk, bv};
  float*       Ps[3]  = {Q, Kd, V};

#pragma unroll
  for (int m = 0; m < 3; ++m) {
    v8f c = {};
#pragma unroll
    for (int kk = 0; kk < 32; ++kk) {
      v2f bf = *(const v2f*)(Ws[m] + (size_t)(oBase + l16) * DIN + kk * 4 + hl * 2);
      c = wmma4(a[kk], bf, c);
    }
    const int   o    = oBase + l16;
    const float bias = Bs[m][o];
    const int   hh   = o >> 4;
    const int   d    = o & 15;
    float* P = Ps[m] + (size_t)(b * H + hh) * N * DK + d;
#pragma unroll
    for (int r = 0; r < 8; ++r) {
      const int n = n0 + r + hl * 8;     // C layout: M = r + 8*hl
      P[(size_t)n * DK] = c[r] + bias;
    }
  }
}

// ---------------------------------------------------------------------------
// Kernel 2: per (b, h, 16-query strip): scores -> bias/mask -> softmax -> attn
// out, out-tile = attn @ V.  Dynamic LDS: 16*SROW score panel + 2048 reduce buf.
// grid = (N/16, H, BS), 256 threads (8 waves).
// ---------------------------------------------------------------------------
__global__ void attn_kernel(const float* __restrict__ Q, const float* __restrict__ Kd,
                            const float* __restrict__ V,
                            const float* __restrict__ sp_enc, const int* __restrict__ adj,
                            float* __restrict__ attn, float* __restrict__ out) {
  extern __shared__ float smem[];
  float* sS   = smem;               // [16][SROW] masked/softmaxed scores
  float* sRed = smem + 16 * SROW;   // [8][256] cross-wave AV reduction

  const int tid  = threadIdx.x;
  const int lane = tid & 31, wave = tid >> 5;
  const int l16  = lane & 15, hl = lane >> 4;
  const int qt = blockIdx.x, h = blockIdx.y, b = blockIdx.z;
  const int bh = b * H + h;
  const int qBase = qt * 16;
  const float* Qp = Q  + (size_t)bh * N * DK;
  const float* Kp = Kd + (size_t)bh * N * DK;
  const float* Vp = V  + (size_t)bh * N * DK;

  // A-fragments of the 16x16 Q tile (K-dim = DK = 16 -> 4 frags)
  v2f aq[4];
#pragma unroll
  for (int kk = 0; kk < 4; ++kk)
    aq[kk] = *(const v2f*)(Qp + (size_t)(qBase + l16) * DK + kk * 4 + hl * 2);

  // ---- Phase 1: scores tiles, wave-strided over 128 key tiles ----
  for (int mt = wave; mt < N / 16; mt += 8) {
    const int mBase = mt * 16;
    if (mt + 8 < N / 16)  // speculative prefetch of next K tile for this wave
      __builtin_prefetch(Kp + (size_t)(mBase + 128) * DK, 0, 1);
    v8f c = {};
#pragma unroll
    for (int kk = 0; kk < 4; ++kk) {
      v2f bf = *(const v2f*)(Kp + (size_t)(mBase + l16) * DK + kk * 4 + hl * 2);
      c = wmma4(aq[kk], bf, c);
    }
    const int m = mBase + l16;
#pragma unroll
    for (int r = 0; r < 8; ++r) {
      const int q = qBase + r + hl * 8;
      float s = c[r] * 0.25f;                       // / sqrt(DK)
      const float sp = sp_enc[(size_t)q * N + m];
      s += (q == m) ? 0.0f : sp;                    // diag-zeroed spatial bias
      if (adj[((size_t)b * N + q) * N + m] == 0) s = -1.0e10f;
      sS[(r + hl * 8) * SROW + m] = s;
    }
  }
  __syncthreads();

  // ---- Phase 2: row softmax (wave w owns rows 2w, 2w+1; half-wave per row) ----
  {
    const int row = wave * 2 + hl;
    float* rp = sS + row * SROW;
    float mx = -3.0e38f;
    for (int j = l16; j < N; j += 16) mx = fmaxf(mx, rp[j]);
#pragma unroll
    for (int off = 8; off > 0; off >>= 1) mx = fmaxf(mx, __shfl_xor(mx, off, 16));
    float sum = 0.0f;
    for (int j = l16; j < N; j += 16) {
      const float e = __expf(rp[j] - mx);
      rp[j] = e;
      sum += e;
    }
#pragma unroll
    for (int off = 8; off > 0; off >>= 1) sum += __shfl_xor(sum, off, 16);
    const float inv = 1.0f / sum;
    float* ap = attn + ((size_t)bh * N + qBase + row) * N;
    for (int j = l16; j < N; j += 16) {
      const float p = rp[j] * inv;
      rp[j] = p;          // keep for AV
      ap[j] = p;          // coalesced attn output
    }
  }
  __syncthreads();

  // ---- Phase 3: out tile = attn(16xN) @ V(Nx16), K-chunks of 4, wave-strided ----
  v8f acc = {};
  for (int ms = wave; ms < N / 4; ms += 8) {
    const int m0 = ms * 4;
    const v2f af = *(const v2f*)(sS + l16 * SROW + m0 + hl * 2);  // A from LDS
    v2f bf;
    bf.x = Vp[(size_t)(m0 + hl * 2) * DK + l16];
    bf.y = Vp[(size_t)(m0 + hl * 2 + 1) * DK + l16];
    acc = wmma4(af, bf, acc);
  }
#pragma unroll
  for (int r = 0; r < 8; ++r) sRed[wave * 256 + r * 32 + lane] = acc[r];
  __syncthreads();
  {
    float v = 0.0f;
#pragma unroll
    for (int w = 0; w < 8; ++w) v += sRed[w * 256 + tid];
    const int laneO = tid & 31, rr = tid >> 5;
    const int M = rr + ((laneO >> 4) << 3);
    const int d = laneO & 15;
    const int n = qBase + M;
    // reference permute(0,2,3,1): feature index = d*H + h
    out[((size_t)b * N + n) * DOUT + d * H + h] = v;
  }
}

// ---------------------------------------------------------------------------
extern "C" void kernel_launch(void* const* d_in, const int* in_sizes, int n_in,
                              void* d_out, int out_size, void* d_ws, size_t ws_size,
                              hipStream_t stream) {
  (void)in_sizes; (void)n_in; (void)out_size; (void)ws_size;
  const float* x      = (const float*)d_in[0];
  const int*   adj    = (const int*)  d_in[1];
  const float* sp_enc = (const float*)d_in[2];
  const float* Wq     = (const float*)d_in[3];
  const float* bq     = (const float*)d_in[4];
  const float* Wk     = (const float*)d_in[5];
  const float* bk     = (const float*)d_in[6];
  const float* Wv     = (const float*)d_in[7];
  const float* bv     = (const float*)d_in[8];

  float* out  = (float*)d_out;                       // (BS, N, DOUT)
  float* attn = out + (size_t)BS * N * DOUT;         // (BS, H, N, N)

  // Workspace: Q, K, V in [b][h][n][dk] layout — 3 * 2 MB = 6 MB
  const size_t QSZ = (size_t)BS * H * N * DK;
  float* Q  = (float*)d_ws;
  float* Kd = Q + QSZ;
  float* V  = Kd + QSZ;

  qkv_proj_kernel<<<dim3(BS * N / 16), 256, 0, stream>>>(
      x, Wq, bq, Wk, bk, Wv, bv, Q, Kd, V);

  const size_t shbytes = (size_t)(16 * SROW + 8 * 256) * sizeof(float); // ~139 KB LDS
  attn_kernel<<<dim3(N / 16, H, BS), 256, shbytes, stream>>>(
      Q, Kd, V, sp_enc, adj, attn, out);
}